// GeometricField_34711925686485
// MI455X (gfx1250) — compile-verified
//
#include <hip/hip_runtime.h>
#include <math.h>

#define NPTS   8192
#define BLOCK  512
#define NWAVE  (BLOCK / 32)       // 16 waves
#define TILE   16                 // rows per block (one WMMA M-tile)
#define JSLICE (NPTS / NWAVE)     // 512 columns per wave

#define TWO_PI_F  6.283185307179586f
#define LAMBDA_A  2.5f
#define REPEL_S   0.15f
#define EPS_R     0.1f
#define NORM_EPSF 1e-8f
#define TAU_W     0.3f

typedef __attribute__((ext_vector_type(2))) float v2f;
typedef __attribute__((ext_vector_type(4))) float v4f;
typedef __attribute__((ext_vector_type(8))) float v8f;

// LDS layout (dynamic shared):
//   float4 jdata[NPTS]           : theta, tau, cos, sin per point   (128 KB)
//   float  parts[NWAVE][TILE][8] : per-wave partial reductions       (8 KB)
//   float  tot[TILE][8]          : combined reductions
//   float  rinv[TILE]            : per-row 1/sum(scores)
#define JD_BYTES     (NPTS * 16)
#define PARTS_FLOATS (NWAVE * TILE * 8)
#define TOT_FLOATS   (TILE * 8)
#define SMEM_BYTES   (JD_BYTES + (PARTS_FLOATS + TOT_FLOATS + TILE) * 4)

// Per-pair score + repulsion magnitude. pi/pj = {theta, tau, cos, sin}.
// Uses raw v_sqrt_f32 / v_rcp_f32 — keep the trans-bound loop on the
// single-instruction transcendental pipe instead of IEEE div expansion.
__device__ __forceinline__ void pair_calc(const float4 pi, int i,
                                          const float4 pj, int j,
                                          float& s, float& inv)
{
    float ad   = fabsf(pi.x - pj.x);
    float dth  = fminf(ad, TWO_PI_F - ad);                 // angular distance
    float dist = fmaf(TAU_W, fabsf(pi.y - pj.y), dth);
    float sc   = __expf(-LAMBDA_A * dist);                 // attention score
    float d    = dth + EPS_R;
    float dx   = pi.z - pj.z;
    float dy   = pi.w - pj.w;
    float n2   = fmaf(dx, dx, dy * dy);
    float norm = __builtin_amdgcn_sqrtf(n2) + NORM_EPSF;
    float iv   = REPEL_S * __builtin_amdgcn_rcpf(d * d * norm);
    bool diag  = (i == j);
    s   = diag ? 0.0f : sc;
    inv = diag ? 0.0f : iv;
}

__global__ __launch_bounds__(BLOCK)
void geo_field_kernel(const float* __restrict__ theta,
                      const float* __restrict__ tau,
                      float* __restrict__ theta_out,
                      float* __restrict__ tau_out,
                      float* __restrict__ attn)
{
    extern __shared__ char smem_raw[];
    float4* jdata  = (float4*)smem_raw;
    float*  parts  = (float*)(smem_raw + JD_BYTES);
    float*  tot    = parts + PARTS_FLOATS;
    float*  rinv_s = tot + TOT_FLOATS;

    const int tid = threadIdx.x;
    const int i0  = blockIdx.x * TILE;

    // ---- Stage all j-point data into LDS ----
    for (int j = tid; j < NPTS; j += BLOCK) {
        float th = theta[j];
        float tu = tau[j];
        jdata[j] = make_float4(th, tu, __cosf(th), __sinf(th));
    }
    __syncthreads();

    const int lane = tid & 31;
    const int wave = tid >> 5;
    const int half = lane >> 4;   // 0: K=0,1 ; 1: K=2,3
    const int idx  = lane & 15;   // row (A) / column (B) index in tile

    const int    irow = i0 + idx;
    const float4 pi   = jdata[irow];

    // Branchless per-lane B-column weights (loop-invariant; selection pattern
    // fixed per lane, only data changes each iteration).
    // B1 columns: 0=cos 1=sin 2=tau 3=1 ; B2 columns: 0=cos 1=sin 2=1
    const float wc  = (idx == 0) ? 1.0f : 0.0f;
    const float ws  = (idx == 1) ? 1.0f : 0.0f;
    const float wt  = (idx == 2) ? 1.0f : 0.0f;
    const float w1  = (idx == 3) ? 1.0f : 0.0f;
    const float w12 = (idx == 2) ? 1.0f : 0.0f;

    // ---- Pass 1: row reductions via fp32 WMMA 16x16x4 ----
    // D1 columns: 0=Σs·cos  1=Σs·sin  2=Σs·tau  3=Σs
    // D2 columns: 0=Σinv·cos 1=Σinv·sin 2=Σinv
    v8f acc1 = {0.f, 0.f, 0.f, 0.f, 0.f, 0.f, 0.f, 0.f};
    v8f acc2 = {0.f, 0.f, 0.f, 0.f, 0.f, 0.f, 0.f, 0.f};

    const int jbase = wave * JSLICE;
    for (int j0 = jbase; j0 < jbase + JSLICE; j0 += 4) {
        const int jA = j0 + half * 2;            // this lane's K pair
        const float4 p0 = jdata[jA];
        const float4 p1 = jdata[jA + 1];

        float s0, iv0, s1, iv1;
        pair_calc(pi, irow, p0, jA,     s0, iv0);
        pair_calc(pi, irow, p1, jA + 1, s1, iv1);

        v2f A1; A1.x = s0;  A1.y = s1;           // A[m=idx][k], k = 2*half + vgpr
        v2f A2; A2.x = iv0; A2.y = iv1;

        // B[k][n=idx] built as a 3-FMA dot with per-lane 0/1 weights (no branches)
        float b0 = fmaf(wc, p0.z, fmaf(ws, p0.w, fmaf(wt, p0.y, w1)));
        float b1 = fmaf(wc, p1.z, fmaf(ws, p1.w, fmaf(wt, p1.y, w1)));
        float c0 = fmaf(wc, p0.z, fmaf(ws, p0.w, w12));
        float c1 = fmaf(wc, p1.z, fmaf(ws, p1.w, w12));
        v2f B1; B1.x = b0; B1.y = b1;
        v2f B2; B2.x = c0; B2.y = c1;

        acc1 = __builtin_amdgcn_wmma_f32_16x16x4_f32(false, A1, false, B1,
                                                     (short)0, acc1, false, false);
        acc2 = __builtin_amdgcn_wmma_f32_16x16x4_f32(false, A2, false, B2,
                                                     (short)0, acc2, false, false);
    }

    // ---- Spill per-wave partial D to LDS (only columns 0..3 are nonzero) ----
    // D layout: VGPR r, lanes 0-15 -> D[r][n], lanes 16-31 -> D[r+8][n]
    if (idx < 4) {
        float* pw = parts + wave * (TILE * 8);
#pragma unroll
        for (int r = 0; r < 8; ++r) {
            const int m = r + 8 * half;
            pw[m * 8 + idx]     = acc1[r];
            pw[m * 8 + 4 + idx] = acc2[r];
        }
    }
    __syncthreads();

    // ---- Combine partials across the 16 waves ----
    if (tid < TILE * 8) {
        const int m = tid >> 3, slot = tid & 7;
        float sum = 0.0f;
        for (int w = 0; w < NWAVE; ++w)
            sum += parts[(w * TILE + m) * 8 + slot];
        tot[m * 8 + slot] = sum;
    }
    __syncthreads();

    // ---- Finalize per-row outputs ----
    if (tid < TILE) {
        const int m = tid;
        const float sum_s = tot[m * 8 + 3];
        const float sc    = tot[m * 8 + 0];
        const float ss    = tot[m * 8 + 1];
        const float st    = tot[m * 8 + 2];
        const float sic   = tot[m * 8 + 4];
        const float sis   = tot[m * 8 + 5];
        const float sinv  = tot[m * 8 + 6];
        const float r     = 1.0f / sum_s;
        const float4 pm   = jdata[i0 + m];
        // force = attn-weighted resultant + factorized repulsion:
        // Σ_j inv·(cos_i - cos_j) = cos_i·Σinv - Σ inv·cos_j
        const float fx = fmaf(sc, r, fmaf(pm.z, sinv, -sic));
        const float fy = fmaf(ss, r, fmaf(pm.w, sinv, -sis));
        theta_out[i0 + m] = atan2f(fy, fx);
        tau_out[i0 + m]   = st * r;
        rinv_s[m] = r;
    }
    __syncthreads();

    // ---- Pass 2: stream normalized attention rows ----
    // Write-once 256 MB output: non-temporal B128 stores, don't pollute caches.
    for (int m = 0; m < TILE; ++m) {
        const int i      = i0 + m;
        const float4 pim = jdata[i];
        const float rr   = rinv_s[m];
        float* row       = attn + (size_t)i * NPTS;
        for (int c0 = tid * 4; c0 < NPTS; c0 += BLOCK * 4) {
            v4f r4;
#pragma unroll
            for (int u = 0; u < 4; ++u) {
                const int col = c0 + u;
                const float4 pj = jdata[col];
                float ad   = fabsf(pim.x - pj.x);
                float dth  = fminf(ad, TWO_PI_F - ad);
                float dist = fmaf(TAU_W, fabsf(pim.y - pj.y), dth);
                float s    = (i == col) ? 0.0f : __expf(-LAMBDA_A * dist);
                r4[u] = s * rr;
            }
            __builtin_nontemporal_store(r4, (v4f*)(row + c0));
        }
    }
}

extern "C" void kernel_launch(void* const* d_in, const int* in_sizes, int n_in,
                              void* d_out, int out_size, void* d_ws, size_t ws_size,
                              hipStream_t stream) {
    const float* theta = (const float*)d_in[0];
    const float* tau   = (const float*)d_in[1];
    float* out       = (float*)d_out;
    float* theta_out = out;                 // [NPTS]
    float* tau_out   = out + NPTS;          // [NPTS]
    float* attn      = out + 2 * NPTS;      // [NPTS * NPTS]

    geo_field_kernel<<<NPTS / TILE, BLOCK, SMEM_BYTES, stream>>>(
        theta, tau, theta_out, tau_out, attn);
}